// RandomSampler_1529008357472
// MI455X (gfx1250) — compile-verified
//
#include <hip/hip_runtime.h>
#include <stdint.h>

#define B_      32
#define C_      64
#define N_      65536
#define K_      8192
#define TK      1024              // k-tile per block
#define THREADS 256               // 8 wave32 waves
#define GROUP   8                 // channels per pipeline stage
#define NGROUP  (C_ / GROUP)      // 8 stages
#define PER_LANE 4                // contiguous k's per lane (b128 store)

__device__ __forceinline__ uint32_t mix32(uint32_t x) {
    x ^= x >> 16; x *= 0x7FEB352Du;
    x ^= x >> 15; x *= 0x846CA68Bu;
    x ^= x >> 16;
    return x;
}

// Keyed bijection on [0, 65536): 4-round Feistel over two 8-bit halves.
// Distinct k -> distinct indices (a true permutation prefix per batch).
__device__ __forceinline__ uint32_t feistel16(uint32_t v, uint32_t key) {
    uint32_t L = (v >> 8) & 0xFFu, R = v & 0xFFu;
#pragma unroll
    for (int r = 0; r < 4; ++r) {
        uint32_t f  = mix32(R + key + (uint32_t)r * 0x9E3779B9u) & 0xFFu;
        uint32_t nL = R;
        R = L ^ f;
        L = nL;
    }
    return (L << 8) | R;
}

// ASYNCcnt-tracked DMA: random 4B gather global->LDS, no VGPR landing pad.
__device__ __forceinline__ void async_gather_b32(uint32_t lds, uint32_t voff,
                                                 const float* base) {
    asm volatile("global_load_async_to_lds_b32 %0, %1, %2"
                 :: "v"(lds), "v"(voff), "s"(base) : "memory");
}

// ASYNCcnt-tracked DMA: coalesced 16B/lane LDS->global stream-out.
__device__ __forceinline__ void async_store_b128(uint32_t voff, uint32_t lds,
                                                 float* base) {
    asm volatile("global_store_async_from_lds_b128 %0, %1, %2"
                 :: "v"(voff), "v"(lds), "s"(base) : "memory");
}

__global__ __launch_bounds__(THREADS)
void sampler_gather(const float* __restrict__ x, float* __restrict__ y) {
    // Two pipeline halves x 8 channels x 1024 floats = 64 KB.
    // Each lane's slots are private to its wave -> no barriers needed.
    __shared__ __align__(16) float stage[2 * GROUP * TK];

    const int b     = blockIdx.x / (K_ / TK);
    const int ktile = blockIdx.x % (K_ / TK);
    const int k0    = ktile * TK;
    const int lane  = threadIdx.x & 31;
    const int wave  = threadIdx.x >> 5;

    // Lane owns 4 CONTIGUOUS k's (16B-aligned) so stores can be b128.
    const int kk = wave * (TK / (THREADS / 32)) + lane * PER_LANE; // 0..TK-4

    const uint32_t key = mix32(0xC0FFEEu ^ (uint32_t)b);
    uint32_t idx4[PER_LANE];                 // gather byte offsets within a row
#pragma unroll
    for (int j = 0; j < PER_LANE; ++j)
        idx4[j] = feistel16((uint32_t)(k0 + kk + j), key) * 4u;

    // Uniform 64-bit bases (SGPR pairs for the saddr forms).
    const float* xb = x + (size_t)b * C_ * (size_t)N_;
    float*       yb = y + (size_t)b * C_ * (size_t)K_;

    // LDS byte address of this lane's slot for (half, channel-in-group).
    auto lds_at = [&](int half, int i) -> uint32_t {
        return (uint32_t)(uintptr_t)&stage[(half * GROUP + i) * TK + kk];
    };

    // Prologue: gather stage 0 (channels 0..7) into half 0.
#pragma unroll
    for (int i = 0; i < GROUP; ++i) {
        const uint32_t cbase = (uint32_t)i * (uint32_t)N_ * 4u;
#pragma unroll
        for (int j = 0; j < PER_LANE; ++j)
            async_gather_b32(lds_at(0, i) + j * 4u, cbase + idx4[j], xb);
    }

    for (int g = 0; g < NGROUP; ++g) {
        const int half  = g & 1;
        const int nhalf = half ^ 1;

        // Drain: gathers for stage g + stream-outs for stage g-1.
        // (One full-latency stall per 8 channels instead of per channel.)
        asm volatile("s_wait_asynccnt 0" ::: "memory");

        // Kick off next stage's 32 gathers into the other half...
        if (g + 1 < NGROUP) {
#pragma unroll
            for (int i = 0; i < GROUP; ++i) {
                const uint32_t cbase =
                    (uint32_t)((g + 1) * GROUP + i) * (uint32_t)N_ * 4u;
#pragma unroll
                for (int j = 0; j < PER_LANE; ++j)
                    async_gather_b32(lds_at(nhalf, i) + j * 4u,
                                     cbase + idx4[j], xb);
            }
        }

        // ...while this stage's 8 channels stream out, 16B per lane.
#pragma unroll
        for (int i = 0; i < GROUP; ++i) {
            const int c = g * GROUP + i;
            const uint32_t voff =
                ((uint32_t)c * (uint32_t)K_ + (uint32_t)(k0 + kk)) * 4u;
            async_store_b128(voff, lds_at(half, i), yb);
        }
    }
    asm volatile("s_wait_asynccnt 0" ::: "memory");
}

extern "C" void kernel_launch(void* const* d_in, const int* in_sizes, int n_in,
                              void* d_out, int out_size, void* d_ws, size_t ws_size,
                              hipStream_t stream) {
    (void)in_sizes; (void)n_in; (void)out_size; (void)d_ws; (void)ws_size;
    const float* x = (const float*)d_in[0];
    float*       y = (float*)d_out;

    dim3 grid(B_ * (K_ / TK));   // 256 blocks: 32 batches x 8 k-tiles
    dim3 block(THREADS);         // 256 threads = 8 waves
    hipLaunchKernelGGL(sampler_gather, grid, block, 0, stream, x, y);
}